// OnnxScatterND_59725815218386
// MI455X (gfx1250) — compile-verified
//
#include <hip/hip_runtime.h>
#include <cstdint>

// ---------------------------------------------------------------------------
// ScatterND: out = data; out[indices] = updates
//   data:    n_rows x 128 f32   (streamed once)
//   updates: m      x 128 f32   (streamed once)
//   indices: m unique row ids   (harness contract: integer -> const int*)
//
// Zero-FLOP, pure-bandwidth problem. Strategy: build row->update map in d_ws
// (stays hot in the 192MB L2), then a single fused pass writes every output
// row from the correct source. All bulk traffic uses b128 vmem with NT
// temporal hints (gfx1250 TH=NT) so the streaming ~1GB does not evict the map.
// ---------------------------------------------------------------------------

typedef __attribute__((ext_vector_type(4))) float v4f;  // native 128-bit vector

#define THREADS 256
#define WAVES_PER_BLOCK (THREADS / 32)
#define ROW_F4 32  // 128 floats per row = 32 x float4

__global__ void init_map_kernel(int* __restrict__ map, int n_rows) {
  int i = blockIdx.x * blockDim.x + threadIdx.x;
  if (i < n_rows) map[i] = -1;
}

__global__ void build_map_kernel(const int* __restrict__ indices,
                                 int* __restrict__ map, int m) {
  int i = blockIdx.x * blockDim.x + threadIdx.x;
  if (i < m) {
    int row = indices[i];
    map[row] = i;  // indices are unique (permutation slice) -> race-free
  }
}

// One wave32 per output row: each lane moves one 128-bit vector
// (global_load_b128 / global_store_b128 with TH=NT).
__global__ void fused_gather_kernel(const v4f* __restrict__ data,
                                    const v4f* __restrict__ updates,
                                    const int* __restrict__ map,
                                    v4f* __restrict__ out,
                                    int n_rows) {
  const int row  = blockIdx.x * WAVES_PER_BLOCK + (threadIdx.x >> 5);
  if (row >= n_rows) return;
  const int lane = threadIdx.x & 31;

  const int u = map[row];  // RT policy: L2-resident, uniform within the wave
  const v4f* __restrict__ src =
      (u >= 0) ? (updates + (size_t)u * ROW_F4)
               : (data    + (size_t)row * ROW_F4);

  v4f v = __builtin_nontemporal_load(src + lane);
  __builtin_nontemporal_store(v, out + (size_t)row * ROW_F4 + lane);
}

// ---- Fallback path (only if workspace is too small for the row map) -------

__global__ void copy_kernel(const v4f* __restrict__ in,
                            v4f* __restrict__ out, long long n4) {
  long long i      = (long long)blockIdx.x * blockDim.x + threadIdx.x;
  long long stride = (long long)gridDim.x * blockDim.x;
  for (; i < n4; i += stride)
    __builtin_nontemporal_store(__builtin_nontemporal_load(in + i), out + i);
}

__global__ void scatter_kernel(const int* __restrict__ indices,
                               const v4f* __restrict__ updates,
                               v4f* __restrict__ out, int m) {
  const int w = blockIdx.x * WAVES_PER_BLOCK + (threadIdx.x >> 5);
  if (w >= m) return;
  const int lane = threadIdx.x & 31;
  const int row  = indices[w];
  v4f v = __builtin_nontemporal_load(updates + (size_t)w * ROW_F4 + lane);
  __builtin_nontemporal_store(v, out + (size_t)row * ROW_F4 + lane);
}

// ---------------------------------------------------------------------------

extern "C" void kernel_launch(void* const* d_in, const int* in_sizes, int n_in,
                              void* d_out, int out_size, void* d_ws, size_t ws_size,
                              hipStream_t stream) {
  const float* data    = (const float*)d_in[0];
  const int*   indices = (const int*)d_in[1];   // harness: integer -> int32
  const float* updates = (const float*)d_in[2];
  float*       out     = (float*)d_out;

  const int D      = 128;
  const int n_rows = in_sizes[0] / D;   // 1,000,000
  const int m      = in_sizes[1];       // 200,000 (flat count of (M,1))

  const v4f* data4    = (const v4f*)data;
  const v4f* updates4 = (const v4f*)updates;
  v4f*       out4     = (v4f*)out;

  if (ws_size >= (size_t)n_rows * sizeof(int)) {
    // Fused path: ~1.03 GB total HBM traffic (~44 us at 23.3 TB/s).
    int* map = (int*)d_ws;

    int g_init = (n_rows + THREADS - 1) / THREADS;
    init_map_kernel<<<g_init, THREADS, 0, stream>>>(map, n_rows);

    int g_build = (m + THREADS - 1) / THREADS;
    build_map_kernel<<<g_build, THREADS, 0, stream>>>(indices, map, m);

    int g_gather = (n_rows + WAVES_PER_BLOCK - 1) / WAVES_PER_BLOCK;
    fused_gather_kernel<<<g_gather, THREADS, 0, stream>>>(
        data4, updates4, map, out4, n_rows);
  } else {
    // Fallback: copy-all then scatter (~1.23 GB traffic), no workspace needed.
    long long n4 = (long long)n_rows * ROW_F4;
    int g_copy = (int)((n4 + THREADS - 1) / THREADS);
    copy_kernel<<<g_copy, THREADS, 0, stream>>>(data4, out4, n4);

    int g_scat = (m + WAVES_PER_BLOCK - 1) / WAVES_PER_BLOCK;
    scatter_kernel<<<g_scat, THREADS, 0, stream>>>(indices, updates4, out4, m);
  }
}